// ModelRPN_44650480009899
// MI455X (gfx1250) — compile-verified
//
#include <hip/hip_runtime.h>
#include <cstdint>
#include <cstddef>

#define B_SZ 8
#define NA 65536
#define NG 128
#define EPSF 1e-4f
#define IOU_SCALEF 10000.0f
#define NEG_TH_ACGTF 3000.0f
#define POS_TH_ACGTF 5000.0f
#define NEG_TH_GTACF 100.0f
#define NUM_POS 128
#define NUM_NEG 128
#define TPB 256
#define NBLK (NA / TPB) /* 256 anchor-blocks per batch */
#define G_TILE 8        /* GTs per block in the gt->ac pass */

// ---------------------------------------------------------------------------
// CDNA5 async global->LDS stage of `n` boxes (16B each) starting at gt_b.
// Lanes 0..n-1 each issue one GLOBAL_LOAD_ASYNC_TO_LDS_B128 (ASYNCcnt path),
// then every wave drains its own ASYNCcnt and the workgroup barriers.
// ---------------------------------------------------------------------------
__device__ __forceinline__ void stage_boxes_lds(const float* __restrict__ gt_b,
                                                float* s_gt, int t, int n) {
    if (t < n) {
        unsigned lds_addr = (unsigned)(unsigned long long)(uintptr_t)&s_gt[t * 4];
        unsigned voff     = (unsigned)(t * 16);
        asm volatile("global_load_async_to_lds_b128 %0, %1, %2"
                     :
                     : "v"(lds_addr), "v"(voff), "s"(gt_b)
                     : "memory");
    }
    asm volatile("s_wait_asynccnt 0" ::: "memory");
    __syncthreads();
}

__device__ __forceinline__ float iou_scaled(float a0, float a1, float a2, float a3,
                                            float area_a,
                                            float g0, float g1, float g2, float g3) {
    float ih    = fminf(a2, g2) - fmaxf(a0, g0);
    float iw    = fminf(a3, g3) - fmaxf(a1, g1);
    float inter = fmaxf(ih, 0.0f) * fmaxf(iw, 0.0f);
    float uni   = area_a + (g2 - g0) * (g3 - g1) - inter;
    return IOU_SCALEF * inter / (uni + EPSF);
}

// K1: per-anchor max/argmax IoU over 128 GTs (GT table in LDS, async-staged).
__global__ void k_iou_acgt(const float* __restrict__ bx_ac,
                           const float* __restrict__ bx_gt,
                           float* __restrict__ ws_iou,
                           uint8_t* __restrict__ ws_idx,
                           uint8_t* __restrict__ ws_forced) {
    __shared__ float s_gt[NG * 4];
    const int t = threadIdx.x;
    const int b = blockIdx.y;
    stage_boxes_lds(bx_gt + (size_t)b * NG * 4, s_gt, t, NG);

    const size_t ai = (size_t)b * NA + (size_t)blockIdx.x * TPB + t;
    const float4 ab = reinterpret_cast<const float4*>(bx_ac)[ai];
    const float area_a = (ab.z - ab.x) * (ab.w - ab.y);

    float best = -1.0f;
    int   bidx = 0;
#pragma unroll 8
    for (int g = 0; g < NG; ++g) {
        float iou = iou_scaled(ab.x, ab.y, ab.z, ab.w, area_a,
                               s_gt[4 * g], s_gt[4 * g + 1], s_gt[4 * g + 2], s_gt[4 * g + 3]);
        if (iou > best) { best = iou; bidx = g; }  // strict '>' == first-max (jnp.argmax)
    }
    ws_iou[ai]    = best;
    ws_idx[ai]    = (uint8_t)bidx;
    ws_forced[ai] = 0xFF;  // init for K2's scatter (stream-ordered)
}

// K2: per-(b, 8-GT tile) max/argmax over all anchors; scatter forced positives.
// Packed (iou_bits<<32)|~anchor: u64 max picks higher IoU, then lowest anchor
// index (== jnp.argmax first-occurrence), independent of traversal order.
// Tiling 8 GTs/block cuts the anchor-slab re-reads 8x vs one-block-per-GT.
__global__ void k_iou_gtac(const float* __restrict__ bx_ac,
                           const float* __restrict__ bx_gt,
                           uint8_t* __restrict__ ws_forced) {
    __shared__ float s_gt[G_TILE * 4];
    __shared__ unsigned long long s_best[G_TILE];
    const int b = blockIdx.y, t = threadIdx.x;
    const int gbase = blockIdx.x * G_TILE;

    if (t < G_TILE) s_best[t] = 0ull;
    stage_boxes_lds(bx_gt + ((size_t)b * NG + gbase) * 4, s_gt, t, G_TILE);

    // GT tile into registers (LDS broadcast reads, once).
    float g0[G_TILE], g1[G_TILE], g2[G_TILE], g3[G_TILE], ag[G_TILE];
#pragma unroll
    for (int k = 0; k < G_TILE; ++k) {
        g0[k] = s_gt[4 * k];     g1[k] = s_gt[4 * k + 1];
        g2[k] = s_gt[4 * k + 2]; g3[k] = s_gt[4 * k + 3];
        ag[k] = (g2[k] - g0[k]) * (g3[k] - g1[k]);
    }

    unsigned long long enc[G_TILE];
#pragma unroll
    for (int k = 0; k < G_TILE; ++k) enc[k] = 0ull;

    const float4* ac4 = reinterpret_cast<const float4*>(bx_ac) + (size_t)b * NA;
    for (int a = t; a < NA; a += TPB) {
        const float4 ab = ac4[a];
        const float area_a = (ab.z - ab.x) * (ab.w - ab.y);
        const unsigned low = ~(unsigned)a;
#pragma unroll
        for (int k = 0; k < G_TILE; ++k) {
            float ih    = fminf(ab.z, g2[k]) - fmaxf(ab.x, g0[k]);
            float iw    = fminf(ab.w, g3[k]) - fmaxf(ab.y, g1[k]);
            float inter = fmaxf(ih, 0.0f) * fmaxf(iw, 0.0f);
            float uni   = area_a + ag[k] - inter;
            float iou   = IOU_SCALEF * inter / (uni + EPSF);
            unsigned long long e =
                ((unsigned long long)__float_as_uint(iou) << 32) | low;
            if (e > enc[k]) enc[k] = e;
        }
    }
#pragma unroll
    for (int k = 0; k < G_TILE; ++k) atomicMax(&s_best[k], enc[k]);  // ds_max_u64
    __syncthreads();
    if (t < G_TILE) {
        const unsigned long long best = s_best[t];
        const float    iou = __uint_as_float((unsigned)(best >> 32));
        const unsigned a   = ~(unsigned)(best & 0xFFFFFFFFull);
        if (iou > NEG_TH_GTACF) ws_forced[(size_t)b * NA + a] = (uint8_t)(gbase + t);
    }
}

// K3: per-block fg/bg counts via wave32 ballots.
__global__ void k_count(const float* __restrict__ ws_iou,
                        const uint8_t* __restrict__ ws_forced,
                        int* __restrict__ fgc, int* __restrict__ bgc) {
    const int t = threadIdx.x, b = blockIdx.y, blk = blockIdx.x;
    const size_t ai = (size_t)b * NA + (size_t)blk * TPB + t;
    const float iou    = ws_iou[ai];
    const bool  forced = ws_forced[ai] != 0xFF;
    const bool  fg     = forced || (iou > POS_TH_ACGTF);
    const bool  bg     = !fg && (iou < NEG_TH_ACGTF);

    __shared__ int wf[TPB / 32], wb[TPB / 32];
    const int lane = t & 31, wid = t >> 5;
    unsigned mf = __builtin_amdgcn_ballot_w32(fg);
    unsigned mb = __builtin_amdgcn_ballot_w32(bg);
    if (lane == 0) { wf[wid] = __popc(mf); wb[wid] = __popc(mb); }
    __syncthreads();
    if (t == 0) {
        int sf = 0, sb = 0;
        for (int i = 0; i < TPB / 32; ++i) { sf += wf[i]; sb += wb[i]; }
        const int bi = b * NBLK + blk;
        fgc[bi] = sf; bgc[bi] = sb;
    }
}

// K4: per-batch exclusive scan of the 256 block counts (Hillis-Steele in LDS).
__global__ void k_scan(const int* __restrict__ fgc, const int* __restrict__ bgc,
                       int* __restrict__ fgb, int* __restrict__ bgb) {
    const int b = blockIdx.x, t = threadIdx.x;
    __shared__ int sf[NBLK], sb[NBLK];
    const int of = fgc[b * NBLK + t], ob = bgc[b * NBLK + t];
    sf[t] = of; sb[t] = ob;
    __syncthreads();
    for (int off = 1; off < NBLK; off <<= 1) {
        int vf = (t >= off) ? sf[t - off] : 0;
        int vb = (t >= off) ? sb[t - off] : 0;
        __syncthreads();
        sf[t] += vf; sb[t] += vb;
        __syncthreads();
    }
    fgb[b * NBLK + t] = sf[t] - of;  // exclusive prefix
    bgb[b * NBLK + t] = sb[t] - ob;
}

// K5: labels (ballot inclusive cumsum + block base) and bbox deltas.
__global__ void k_final(const float* __restrict__ bx_ac,
                        const float* __restrict__ bx_gt,
                        const float* __restrict__ ws_iou,
                        const uint8_t* __restrict__ ws_idx,
                        const uint8_t* __restrict__ ws_forced,
                        const int* __restrict__ fgb, const int* __restrict__ bgb,
                        float* __restrict__ out_del, float* __restrict__ out_lab) {
    __shared__ float s_gt[NG * 4];
    __shared__ int wf[TPB / 32], wb[TPB / 32];
    const int t = threadIdx.x, b = blockIdx.y, blk = blockIdx.x;
    stage_boxes_lds(bx_gt + (size_t)b * NG * 4, s_gt, t, NG);

    const size_t  ai     = (size_t)b * NA + (size_t)blk * TPB + t;
    const float   iou    = ws_iou[ai];
    const uint8_t fidx   = ws_forced[ai];
    const bool    forced = fidx != 0xFF;
    const bool    fg     = forced || (iou > POS_TH_ACGTF);
    const bool    bg     = !fg && (iou < NEG_TH_ACGTF);

    const int lane = t & 31, wid = t >> 5;
    unsigned mf = __builtin_amdgcn_ballot_w32(fg);
    unsigned mb = __builtin_amdgcn_ballot_w32(bg);
    const unsigned lm = (1u << lane) - 1u;
    const int lpf = __popc(mf & lm), lpb = __popc(mb & lm);
    if (lane == 0) { wf[wid] = __popc(mf); wb[wid] = __popc(mb); }
    __syncthreads();
    int wbf = 0, wbb = 0;
    for (int i = 0; i < wid; ++i) { wbf += wf[i]; wbb += wb[i]; }
    const int bi = b * NBLK + blk;
    const int cf = fgb[bi] + wbf + lpf + (fg ? 1 : 0);  // inclusive cumsum(fg)
    const int cb = bgb[bi] + wbb + lpb + (bg ? 1 : 0);  // inclusive cumsum(bg)
    const bool sp = fg && (cf <= NUM_POS);
    const bool sn = bg && (cb <= NUM_NEG);
    out_lab[ai] = sp ? 1.0f : (sn ? 0.0f : -1.0f);

    float4 d = make_float4(0.0f, 0.0f, 0.0f, 0.0f);
    if (fg) {
        const float4 ab = reinterpret_cast<const float4*>(bx_ac)[ai];
        const int g  = forced ? (int)fidx : (int)ws_idx[ai];
        const float l0 = s_gt[4 * g], l1 = s_gt[4 * g + 1];
        const float l2 = s_gt[4 * g + 2], l3 = s_gt[4 * g + 3];
        const float hr  = fmaxf(ab.z - ab.x, EPSF);
        const float wr  = fmaxf(ab.w - ab.y, EPSF);
        const float ycr = ab.x + 0.5f * hr, xcr = ab.y + 0.5f * wr;
        const float hl  = l2 - l0, wl = l3 - l1;
        const float ycl = l0 + 0.5f * hl, xcl = l1 + 0.5f * wl;
        d.x = fminf(fmaxf((xcl - xcr) / wr, -999.0f), 999.0f);
        d.y = fminf(fmaxf((ycl - ycr) / hr, -999.0f), 999.0f);
        d.z = fminf(fmaxf(logf(wl / wr),   -999.0f), 999.0f);
        d.w = fminf(fmaxf(logf(hl / hr),   -999.0f), 999.0f);
    }
    reinterpret_cast<float4*>(out_del)[ai] = d;
}

extern "C" void kernel_launch(void* const* d_in, const int* in_sizes, int n_in,
                              void* d_out, int out_size, void* d_ws, size_t ws_size,
                              hipStream_t stream) {
    (void)in_sizes; (void)n_in; (void)out_size; (void)ws_size;
    const float* bx_ac = (const float*)d_in[0];
    const float* bx_gt = (const float*)d_in[1];

    char* ws = (char*)d_ws;
    float*   ws_iou    = (float*)ws;   ws += (size_t)B_SZ * NA * sizeof(float);
    uint8_t* ws_idx    = (uint8_t*)ws; ws += (size_t)B_SZ * NA;
    uint8_t* ws_forced = (uint8_t*)ws; ws += (size_t)B_SZ * NA;
    int* fgc = (int*)ws; ws += (size_t)B_SZ * NBLK * sizeof(int);
    int* bgc = (int*)ws; ws += (size_t)B_SZ * NBLK * sizeof(int);
    int* fgb = (int*)ws; ws += (size_t)B_SZ * NBLK * sizeof(int);
    int* bgb = (int*)ws;

    float* out_del = (float*)d_out;
    float* out_lab = out_del + (size_t)B_SZ * NA * 4;

    const dim3 gA(NBLK, B_SZ);          // 256 x 8 blocks, 256 threads (8 waves)
    const dim3 gG(NG / G_TILE, B_SZ);   // 16 x 8 blocks, 8 GTs each

    k_iou_acgt<<<gA, TPB, 0, stream>>>(bx_ac, bx_gt, ws_iou, ws_idx, ws_forced);
    k_iou_gtac<<<gG, TPB, 0, stream>>>(bx_ac, bx_gt, ws_forced);
    k_count  <<<gA, TPB, 0, stream>>>(ws_iou, ws_forced, fgc, bgc);
    k_scan   <<<B_SZ, NBLK, 0, stream>>>(fgc, bgc, fgb, bgb);
    k_final  <<<gA, TPB, 0, stream>>>(bx_ac, bx_gt, ws_iou, ws_idx, ws_forced,
                                      fgb, bgb, out_del, out_lab);
}